// GINWrapper_70987219469126
// MI455X (gfx1250) — compile-verified
//
#include <hip/hip_runtime.h>
#include <hip/hip_bf16.h>

#define N_NODES 10000
#define N_EDGES 640000
#define DIM     128
#define LSTRIDE 132   // LDS row stride in floats (pad to spread 64 banks)

typedef __attribute__((ext_vector_type(2))) float v2f;
typedef __attribute__((ext_vector_type(8))) float v8f;

// ---------------------------------------------------------------------------
// Kernel 1: h_pre[i,:] = (1 + eps) * x[i,:]
// ---------------------------------------------------------------------------
__global__ void gin_init_kernel(const float* __restrict__ x,
                                const float* __restrict__ eps_p,
                                float* __restrict__ h_pre) {
    const float s = 1.0f + eps_p[0];
    int gid = blockIdx.x * blockDim.x + threadIdx.x;   // one float4 per thread
    if (gid < (N_NODES * DIM) / 4) {
        float4 v = ((const float4*)x)[gid];
        v.x *= s; v.y *= s; v.z *= s; v.w *= s;
        ((float4*)h_pre)[gid] = v;
    }
}

// ---------------------------------------------------------------------------
// Kernel 2: wave-per-edge scatter-add: h_pre[dst,:] += x[src,:]
// Each lane handles 4 consecutive features (lane*4 .. lane*4+3).
// x and h_pre are L2-resident (5 MB each) so gather + atomics stay on-die.
// ---------------------------------------------------------------------------
__global__ void gin_scatter_kernel(const float* __restrict__ x,
                                   const long long* __restrict__ edge_index,
                                   float* __restrict__ h_pre) {
    int gid  = blockIdx.x * blockDim.x + threadIdx.x;
    int e    = gid >> 5;          // edge id (wave per edge)
    int lane = gid & 31;          // feature chunk
    if (e >= N_EDGES) return;
    int src = (int)edge_index[e];
    int dst = (int)edge_index[N_EDGES + e];
    float4 v = ((const float4*)(x + (size_t)src * DIM))[lane];
    float* d = h_pre + (size_t)dst * DIM + lane * 4;
    atomicAdd(d + 0, v.x);
    atomicAdd(d + 1, v.y);
    atomicAdd(d + 2, v.z);
    atomicAdd(d + 3, v.w);
}

// ---------------------------------------------------------------------------
// Kernel 3: out = relu( relu(h_pre @ W1 + b1) @ W2 + b2 )
// Block = 256 threads = 8 waves; block owns a 16-row stripe (M-tile),
// wave w owns N-columns [16w, 16w+16). Full fp32 via V_WMMA_F32_16X16X4_F32.
// ---------------------------------------------------------------------------
__global__ void __launch_bounds__(256)
gin_mlp_kernel(const float* __restrict__ h_pre,
               const float* __restrict__ W1, const float* __restrict__ b1,
               const float* __restrict__ W2, const float* __restrict__ b2,
               float* __restrict__ out) {
    __shared__ float sh_a[16 * LSTRIDE];   // A stripe for GEMM1 (h_pre tile)
    __shared__ float sh_t[16 * LSTRIDE];   // intermediate T1 tile

    const int tid   = threadIdx.x;
    const int wave  = tid >> 5;
    const int lane  = tid & 31;
    const int lhalf = lane >> 4;     // 0: lanes 0-15, 1: lanes 16-31
    const int lmod  = lane & 15;
    const int m0    = blockIdx.x * 16;     // 625 blocks * 16 = 10000 exactly
    const int n0    = wave * 16;

    // --- cooperative load of the 16x128 A stripe into LDS (float4) ---
    for (int i = tid; i < 16 * (DIM / 4); i += 256) {
        int r  = i >> 5;            // row 0..15
        int c4 = i & 31;            // float4 column
        float4 v = ((const float4*)(h_pre + (size_t)(m0 + r) * DIM))[c4];
        float* p = &sh_a[r * LSTRIDE + c4 * 4];
        p[0] = v.x; p[1] = v.y; p[2] = v.z; p[3] = v.w;
    }
    __syncthreads();

    // --- GEMM1: T1 = relu(A @ W1 + b1) ---
    v8f acc = {};
    #pragma unroll 4
    for (int k0 = 0; k0 < DIM; k0 += 4) {
        const int ka = k0 + 2 * lhalf;
        v2f a, b;
        a.x = sh_a[lmod * LSTRIDE + ka];
        a.y = sh_a[lmod * LSTRIDE + ka + 1];
        b.x = W1[(size_t)ka * DIM + n0 + lmod];
        b.y = W1[(size_t)(ka + 1) * DIM + n0 + lmod];
        acc = __builtin_amdgcn_wmma_f32_16x16x4_f32(false, a, false, b,
                                                    (short)0, acc, false, false);
    }
    {
        const float bias = b1[n0 + lmod];
        #pragma unroll
        for (int v = 0; v < 8; ++v) {
            int r = v + 8 * lhalf;
            sh_t[r * LSTRIDE + n0 + lmod] = fmaxf(acc[v] + bias, 0.0f);
        }
    }
    __syncthreads();

    // --- GEMM2: out = relu(T1 @ W2 + b2) ---
    v8f acc2 = {};
    #pragma unroll 4
    for (int k0 = 0; k0 < DIM; k0 += 4) {
        const int ka = k0 + 2 * lhalf;
        v2f a, b;
        a.x = sh_t[lmod * LSTRIDE + ka];
        a.y = sh_t[lmod * LSTRIDE + ka + 1];
        b.x = W2[(size_t)ka * DIM + n0 + lmod];
        b.y = W2[(size_t)(ka + 1) * DIM + n0 + lmod];
        acc2 = __builtin_amdgcn_wmma_f32_16x16x4_f32(false, a, false, b,
                                                     (short)0, acc2, false, false);
    }
    {
        const float bias = b2[n0 + lmod];
        #pragma unroll
        for (int v = 0; v < 8; ++v) {
            int r = v + 8 * lhalf;
            out[(size_t)(m0 + r) * DIM + n0 + lmod] = fmaxf(acc2[v] + bias, 0.0f);
        }
    }
}

// ---------------------------------------------------------------------------
// Launch: inputs in order {x, eps, W1, b1, W2, b2, edge_index}
// ---------------------------------------------------------------------------
extern "C" void kernel_launch(void* const* d_in, const int* in_sizes, int n_in,
                              void* d_out, int out_size, void* d_ws, size_t ws_size,
                              hipStream_t stream) {
    const float*     x    = (const float*)d_in[0];
    const float*     eps  = (const float*)d_in[1];
    const float*     W1   = (const float*)d_in[2];
    const float*     b1   = (const float*)d_in[3];
    const float*     W2   = (const float*)d_in[4];
    const float*     b2   = (const float*)d_in[5];
    const long long* ei   = (const long long*)d_in[6];
    float*           out  = (float*)d_out;
    float*           h_pre = (float*)d_ws;            // N_NODES*DIM floats

    // 1) seed accumulator with (1+eps)*x
    {
        int n4 = (N_NODES * DIM) / 4;
        int blocks = (n4 + 255) / 256;
        gin_init_kernel<<<blocks, 256, 0, stream>>>(x, eps, h_pre);
    }
    // 2) edge scatter-add (wave per edge)
    {
        long long threads = (long long)N_EDGES * 32;
        int blocks = (int)((threads + 255) / 256);
        gin_scatter_kernel<<<blocks, 256, 0, stream>>>(x, ei, h_pre);
    }
    // 3) fused MLP with fp32 WMMA
    {
        gin_mlp_kernel<<<N_NODES / 16, 256, 0, stream>>>(h_pre, W1, b1, W2, b2, out);
    }
}